// SuperMegnetGlobalModel_47974784696400
// MI455X (gfx1250) — compile-verified
//
#include <hip/hip_runtime.h>
#include <cmath>

// CDNA5 / gfx1250: wave32, WMMA fp32 16x16x4.
typedef __attribute__((ext_vector_type(2))) float v2f;
typedef __attribute__((ext_vector_type(4))) float v4f;
typedef __attribute__((ext_vector_type(8))) float v8f;

#define HID    160
#define GROUPS 10
#define GSIZE  16          // HID / GROUPS == WMMA tile N-width (key fusion fact)
#define NT     (HID / 16)  // 10 N-tiles per row block
#define EPS_GN 1e-5f

__device__ __forceinline__ float softplus_f(float x) {
    // numerically stable softplus, matches jax.nn.softplus
    if (x > 20.0f)  return x;
    if (x < -20.0f) return __expf(x);
    return log1pf(__expf(x));
}

// ---------------------------------------------------------------------------
// Kernel 1: streaming segment-sum — the bandwidth-dominant part (~1 GB read
// once => ~44us floor at 23.3 TB/s). batch_idx is sorted, so segments are
// contiguous: no atomics, pure streaming. One block per (graph, tensor).
//
// Thread layout: 160 threads = 4 row-slots x 40 float4-columns. Each thread
// streams 16B non-temporal b128 loads (row stride 640B keeps 16B alignment),
// 2-deep unrolled, then a small LDS reduction folds the 4 row-slots.
// NT loads keep this single-use 1 GB stream from thrashing L2.
// ---------------------------------------------------------------------------
__global__ void seg_reduce_kernel(const float* __restrict__ node,
                                  const float* __restrict__ edge,
                                  const int*   __restrict__ bidx,
                                  float* __restrict__ node_aggr,
                                  float* __restrict__ edge_aggr,
                                  int N, int B) {
    const int blk = blockIdx.x;
    const int b   = (blk < B) ? blk : (blk - B);
    const float* __restrict__ src = (blk < B) ? node : edge;
    float* __restrict__ dst       = (blk < B) ? node_aggr : edge_aggr;

    const int t  = threadIdx.x;   // 0..159
    const int r  = t / 40;        // row slot 0..3
    const int c4 = (t % 40) * 4;  // float4 column start (0..156)

    // lower_bound(bidx, b) and upper_bound(bidx, b) on the sorted index.
    // b is block-uniform -> compiler scalarizes this to SALU/SMEM.
    int lo = 0, hi = N;
    while (lo < hi) { int mid = (lo + hi) >> 1; if (bidx[mid] <  b) lo = mid + 1; else hi = mid; }
    const int start = lo;
    hi = N;
    while (lo < hi) { int mid = (lo + hi) >> 1; if (bidx[mid] <= b) lo = mid + 1; else hi = mid; }
    const int end = lo;

    v4f acc = {};
    int row = start + r;                       // this thread's rows: step 4
    for (; row + 4 < end; row += 8) {          // 2 b128 NT loads in flight
        v4f a0 = __builtin_nontemporal_load(
            (const v4f*)&src[(size_t)row * HID + c4]);
        v4f a1 = __builtin_nontemporal_load(
            (const v4f*)&src[(size_t)(row + 4) * HID + c4]);
        acc += a0 + a1;
    }
    if (row < end)
        acc += __builtin_nontemporal_load(
            (const v4f*)&src[(size_t)row * HID + c4]);

    // fold the 4 row-slots via LDS, then one b128 store per aggregate row
    __shared__ v4f sm[160];
    sm[t] = acc;
    __syncthreads();
    if (t < 40) {
        v4f s = sm[t] + sm[t + 40] + sm[t + 80] + sm[t + 120];
        *(v4f*)&dst[(size_t)b * HID + t * 4] = s;
    }
}

// ---------------------------------------------------------------------------
// WMMA fragment loaders (fp32 16x16x4).
// A (16x4, row-major src): lanes 0-15 -> K={k,k+1}; lanes 16-31 -> K={k+2,k+3}
// B (4x16,  row-major src): VGPR0 = row k (lanes 0-15) / k+1 (lanes 16-31);
//                           VGPR1 = row k+2 / k+3
// ---------------------------------------------------------------------------
__device__ __forceinline__ v2f load_a_frag(const float* __restrict__ A,
                                           int m0, int k, int half, int l) {
    const float* p = A + (size_t)(m0 + l) * HID + (k + 2 * half);
    v2f a; a.x = p[0]; a.y = p[1];
    return a;
}

__device__ __forceinline__ v2f load_b_frag(const float* __restrict__ Bm, int ldb,
                                           int n0, int k, int half, int l) {
    const float* p = Bm + (size_t)(k + half) * ldb + (n0 + l);
    v2f b; b.x = p[0]; b.y = p[2 * ldb];   // rows k+half and k+2+half
    return b;
}

__device__ __forceinline__ v8f wmma_k4(v8f c, v2f a, v2f b) {
    return __builtin_amdgcn_wmma_f32_16x16x4_f32(false, a, false, b,
                                                 (short)0, c, false, false);
}

// ---------------------------------------------------------------------------
// Kernel 2: out0 = softplus([global | node_aggr | edge_aggr] @ Wc + bc)
// The 3H concat is never materialized: K runs over three 160-wide sources.
// One wave (32 threads) per 16x16 output tile; EXEC all-1s for WMMA.
// ---------------------------------------------------------------------------
__global__ void combine_kernel(const float* __restrict__ gfeat,
                               const float* __restrict__ naggr,
                               const float* __restrict__ eaggr,
                               const float* __restrict__ Wc,   // [3*HID, HID]
                               const float* __restrict__ bc,
                               float* __restrict__ out0) {
    const int tile = blockIdx.x;
    const int m0   = (tile / NT) * 16;
    const int n0   = (tile % NT) * 16;
    const int lane = threadIdx.x;
    const int half = lane >> 4;
    const int l    = lane & 15;

    v8f c = {};
    const float* srcs[3] = { gfeat, naggr, eaggr };
    #pragma unroll
    for (int s = 0; s < 3; ++s) {
        const float* A  = srcs[s];
        const float* Bm = Wc + (size_t)s * HID * HID;
        for (int k = 0; k < HID; k += 4) {
            v2f a = load_a_frag(A, m0, k, half, l);
            v2f b = load_b_frag(Bm, HID, n0, k, half, l);
            c = wmma_k4(c, a, b);
        }
    }

    const int   n    = n0 + l;
    const float bias = bc[n];
    const int   mb   = m0 + (half << 3);
    #pragma unroll
    for (int r = 0; r < 8; ++r)
        out0[(size_t)(mb + r) * HID + n] = softplus_f(c[r] + bias);
}

// ---------------------------------------------------------------------------
// Kernel 3: H = softplus(X @ W + b)   (M=1024, K=N=160)
// ---------------------------------------------------------------------------
__global__ void mlp1_kernel(const float* __restrict__ X,
                            const float* __restrict__ W,   // [HID, HID]
                            const float* __restrict__ bias,
                            float* __restrict__ Hout) {
    const int tile = blockIdx.x;
    const int m0   = (tile / NT) * 16;
    const int n0   = (tile % NT) * 16;
    const int lane = threadIdx.x;
    const int half = lane >> 4;
    const int l    = lane & 15;

    v8f c = {};
    for (int k = 0; k < HID; k += 4) {
        v2f a = load_a_frag(X, m0, k, half, l);
        v2f b = load_b_frag(W, HID, n0, k, half, l);
        c = wmma_k4(c, a, b);
    }

    const int   n  = n0 + l;
    const float bv = bias[n];
    const int   mb = m0 + (half << 3);
    #pragma unroll
    for (int r = 0; r < 8; ++r)
        Hout[(size_t)(mb + r) * HID + n] = softplus_f(c[r] + bv);
}

// ---------------------------------------------------------------------------
// Kernel 4: dst = prev + GroupNorm(H @ W + b) * gamma + beta  [+ extra]
// Group size (16) == tile N-width, so each tile's 16 lanes per EXEC-half hold
// exactly one (row, group): reduce mean/var with shfl_xor masks 1,2,4,8
// (stays inside each 16-lane half). 'extra' (may be null) folds in the final
// "+ initial" of the reference on the last block.
// ---------------------------------------------------------------------------
__global__ void mlp2_gn_kernel(const float* __restrict__ Hin,
                               const float* __restrict__ W,
                               const float* __restrict__ bias,
                               const float* __restrict__ gamma,
                               const float* __restrict__ beta,
                               const float* __restrict__ prev,
                               const float* __restrict__ extra,
                               float* __restrict__ dst) {
    const int tile = blockIdx.x;
    const int m0   = (tile / NT) * 16;
    const int n0   = (tile % NT) * 16;
    const int lane = threadIdx.x;
    const int half = lane >> 4;
    const int l    = lane & 15;

    v8f c = {};
    for (int k = 0; k < HID; k += 4) {
        v2f a = load_a_frag(Hin, m0, k, half, l);
        v2f b = load_b_frag(W, HID, n0, k, half, l);
        c = wmma_k4(c, a, b);
    }

    const int   n  = n0 + l;
    const float bv = bias[n];
    const float gm = gamma[n];
    const float bt = beta[n];
    const int   mb = m0 + (half << 3);

    #pragma unroll
    for (int r = 0; r < 8; ++r) {
        float v  = c[r] + bv;
        float s  = v;
        float ss = v * v;
        #pragma unroll
        for (int m = 1; m < GSIZE; m <<= 1) {   // reduce within 16-lane half
            s  += __shfl_xor(s,  m, 32);
            ss += __shfl_xor(ss, m, 32);
        }
        const float mean = s * (1.0f / GSIZE);
        const float var  = ss * (1.0f / GSIZE) - mean * mean;   // biased, like jnp.var
        const float xn   = (v - mean) * rsqrtf(var + EPS_GN);

        const size_t idx = (size_t)(mb + r) * HID + n;
        float o = prev[idx] + xn * gm + bt;
        if (extra) o += extra[idx];
        dst[idx] = o;
    }
}

// ---------------------------------------------------------------------------
// Launch: 6 launches (4 kernels), all on `stream`; scratch lives in d_ws (5 * B*HID fp32).
// ---------------------------------------------------------------------------
extern "C" void kernel_launch(void* const* d_in, const int* in_sizes, int n_in,
                              void* d_out, int out_size, void* d_ws, size_t ws_size,
                              hipStream_t stream) {
    const float* gfeat = (const float*)d_in[0];   // [B, HID]
    const float* node  = (const float*)d_in[1];   // [N, HID]
    const float* edge  = (const float*)d_in[2];   // [N, HID]
    const int*   bidx  = (const int*)  d_in[3];   // [N], sorted
    const float* Wc    = (const float*)d_in[4];   // [3H, H]
    const float* bc    = (const float*)d_in[5];
    const float* W1a   = (const float*)d_in[6];
    const float* b1a   = (const float*)d_in[7];
    const float* W2a   = (const float*)d_in[8];
    const float* b2a   = (const float*)d_in[9];
    const float* g1    = (const float*)d_in[10];
    const float* be1   = (const float*)d_in[11];
    const float* W1b   = (const float*)d_in[12];
    const float* b1b   = (const float*)d_in[13];
    const float* W2b   = (const float*)d_in[14];
    const float* b2b   = (const float*)d_in[15];
    const float* g2    = (const float*)d_in[16];
    const float* be2   = (const float*)d_in[17];

    const int B = in_sizes[0] / HID;   // 1024
    const int N = in_sizes[1] / HID;   // 800000

    float* ws    = (float*)d_ws;
    const size_t mat = (size_t)B * HID;
    float* naggr = ws;             // [B, HID]
    float* eaggr = naggr + mat;    // [B, HID]
    float* out0  = eaggr + mat;    // initial (combine output)
    float* out1  = out0  + mat;    // after block 1
    float* hbuf  = out1  + mat;    // MLP hidden

    // 1) bandwidth-dominant segment sums (contiguous segments, b128 NT stream)
    seg_reduce_kernel<<<2 * B, HID, 0, stream>>>(node, edge, bidx, naggr, eaggr, N, B);

    const int tiles = (B / 16) * NT;  // 64 * 10 = 640 waves

    // 2) out0 = softplus(concat @ Wc + bc)     (== "initial")
    combine_kernel<<<tiles, 32, 0, stream>>>(gfeat, naggr, eaggr, Wc, bc, out0);

    // 3) block 1: out1 = out0 + GN(softplus(out0@W1a+b1a)@W2a+b2a)
    mlp1_kernel  <<<tiles, 32, 0, stream>>>(out0, W1a, b1a, hbuf);
    mlp2_gn_kernel<<<tiles, 32, 0, stream>>>(hbuf, W2a, b2a, g1, be1, out0, nullptr, out1);

    // 4) block 2 + final "+ initial" folded into the epilogue:
    //    d_out = out1 + GN(softplus(out1@W1b+b1b)@W2b+b2b) + out0
    mlp1_kernel  <<<tiles, 32, 0, stream>>>(out1, W1b, b1b, hbuf);
    mlp2_gn_kernel<<<tiles, 32, 0, stream>>>(hbuf, W2b, b2b, g2, be2, out1, out0,
                                             (float*)d_out);
}